// STBlock_24335284699211
// MI455X (gfx1250) — compile-verified
//
#include <hip/hip_runtime.h>
#include <hip/hip_bf16.h>

// ---------------- problem constants ----------------
#define BB     8
#define NN_    1024
#define FIN    32
#define FOUT   64
#define TT     24
#define EE     16384
#define BT     (BB*TT)            // 192
#define MFLAT  (NN_*FIN)          // 32768
#define D_     (FIN*TT)           // 768
#define XB     ((long)MFLAT*TT)   // per-batch x elements: 786432

typedef __attribute__((ext_vector_type(2))) float v2f;
typedef __attribute__((ext_vector_type(8))) float v8f;

__device__ __forceinline__ float sigmoidf_(float x) { return 1.0f / (1.0f + __expf(-x)); }

// ================= Phase 1: temporal attention =================

// score_t[b,t,u] = sum_m x[b,m,t]*x[b,m,u]   (m over N*F = 32768)
__global__ void temporal_score(const float* __restrict__ x, float* __restrict__ scoreT) {
    __shared__ float tile[256 * TT];          // 24 KB
    const int b   = blockIdx.x;
    const int tid = threadIdx.x;              // 0..575
    const int t = tid / TT, u = tid % TT;
    const float* xb = x + (long)b * XB;
    float acc = 0.f;
    for (int c = 0; c < MFLAT / 256; ++c) {
        for (int i = tid; i < 256 * TT; i += 576) tile[i] = xb[(long)c * (256 * TT) + i];
        __syncthreads();
        for (int mm = 0; mm < 256; ++mm) acc += tile[mm * TT + t] * tile[mm * TT + u];
        __syncthreads();
    }
    scoreT[b * (TT * TT) + tid] = acc;
}

// E_att[b,t,:] = softmax_v( sum_u Ve[t,u]*sigmoid(score_t[b,u,v]) + be[t,v] )
__global__ void temporal_att(const float* __restrict__ scoreT, const float* __restrict__ Ve,
                             const float* __restrict__ be, float* __restrict__ Eatt) {
    const int tid = threadIdx.x;              // 0..191
    const int b = tid / TT, t = tid % TT;
    float rowv[TT];
    float mx = -3.0e38f;
    for (int v = 0; v < TT; ++v) {
        float s = be[t * TT + v];
        for (int u = 0; u < TT; ++u)
            s += Ve[t * TT + u] * sigmoidf_(scoreT[b * (TT * TT) + u * TT + v]);
        rowv[v] = s;
        mx = fmaxf(mx, s);
    }
    float sum = 0.f;
    for (int v = 0; v < TT; ++v) { rowv[v] = __expf(rowv[v] - mx); sum += rowv[v]; }
    const float inv = 1.0f / sum;
    for (int v = 0; v < TT; ++v) Eatt[b * (TT * TT) + t * TT + v] = rowv[v] * inv;
}

// x_TA[b,m,t] = sum_u x_flat[b,m,u] * E_att[b,u,t]
__global__ void apply_temporal(const float* __restrict__ x, const float* __restrict__ Eatt,
                               float* __restrict__ xta) {
    __shared__ float E[TT * TT];
    const int tid = threadIdx.x;
    const long g = (long)blockIdx.x * 256 + tid;      // over B*MFLAT = 262144
    const int b = (int)(g >> 15);                      // 32768 rows per batch
    for (int i = tid; i < TT * TT; i += 256) E[i] = Eatt[b * (TT * TT) + i];
    __syncthreads();
    const float* xr = x + g * TT;
    float xv[TT];
#pragma unroll
    for (int u = 0; u < TT; ++u) xv[u] = xr[u];
    float* o = xta + g * TT;
#pragma unroll
    for (int t = 0; t < TT; ++t) {
        float s = 0.f;
#pragma unroll
        for (int u = 0; u < TT; ++u) s += xv[u] * E[u * TT + t];
        o[t] = s;
    }
}

// ================= WMMA fp32 GEMM =================
// Each wave computes a 32x32 C block (2x2 of 16x16 WMMA tiles, register blocked):
// per K-step of 4 -> 2 A-frag + 2 B-frag loads feed 4 v_wmma_f32_16x16x4_f32.
// Block = 128 threads (4 waves), each wave a different 32-row strip.
// C[m,n] = epilogue( sum_k A[m,k]*B[k,n] )   TRANSB: B given as (N,K) row-major
// EPI: 0 = sigmoid, 1 = + bias[m*ldc+n], 2 = none
template <int TRANSB, int EPI>
__global__ void wmma_gemm_f32(const float* __restrict__ A, const float* __restrict__ B,
                              float* __restrict__ C, const float* __restrict__ bias,
                              int lda, int ldb, int ldc, int Kdim,
                              long strideA, long strideB, long strideC) {
    const int tid  = threadIdx.x;
    const int wave = tid >> 5;
    const int lane = tid & 31;
    const int half = lane >> 4;      // 0: K 0/1, 1: K 2/3
    const int r    = lane & 15;
    const int tileM = (blockIdx.x * 4 + wave) * 32;
    const int tileN = blockIdx.y * 32;
    const int b     = blockIdx.z;
    const float* Ab = A + (long)b * strideA;
    const float* Bb = B + (long)b * strideB;
    float*       Cb = C + (long)b * strideC;

    const int m0 = tileM + r, m1 = m0 + 16;
    const int n0 = tileN + r, n1 = n0 + 16;

    v8f acc00 = {}, acc01 = {}, acc10 = {}, acc11 = {};
#pragma unroll 4
    for (int k = 0; k < Kdim; k += 4) {
        const int kk = k + 2 * half;
        v2f a0, a1, b0, b1;
        a0.x = Ab[(long)m0 * lda + kk];
        a0.y = Ab[(long)m0 * lda + kk + 1];
        a1.x = Ab[(long)m1 * lda + kk];
        a1.y = Ab[(long)m1 * lda + kk + 1];
        if (TRANSB) {
            b0.x = Bb[(long)n0 * ldb + kk];
            b0.y = Bb[(long)n0 * ldb + kk + 1];
            b1.x = Bb[(long)n1 * ldb + kk];
            b1.y = Bb[(long)n1 * ldb + kk + 1];
        } else {
            b0.x = Bb[(long)kk * ldb + n0];
            b0.y = Bb[(long)(kk + 1) * ldb + n0];
            b1.x = Bb[(long)kk * ldb + n1];
            b1.y = Bb[(long)(kk + 1) * ldb + n1];
        }
        acc00 = __builtin_amdgcn_wmma_f32_16x16x4_f32(false, a0, false, b0, (short)0, acc00, false, false);
        acc01 = __builtin_amdgcn_wmma_f32_16x16x4_f32(false, a0, false, b1, (short)0, acc01, false, false);
        acc10 = __builtin_amdgcn_wmma_f32_16x16x4_f32(false, a1, false, b0, (short)0, acc10, false, false);
        acc11 = __builtin_amdgcn_wmma_f32_16x16x4_f32(false, a1, false, b1, (short)0, acc11, false, false);
    }

    // epilogue + store: C/D layout: VGPR v -> row (v + 8*half), col = lane&15
    const int colA = tileN + r, colB = tileN + 16 + r;
#pragma unroll
    for (int v = 0; v < 8; ++v) {
        const int rowA = tileM + v + half * 8;
        const int rowB = rowA + 16;
        float v00 = acc00[v], v01 = acc01[v], v10 = acc10[v], v11 = acc11[v];
        if (EPI == 0) {
            v00 = sigmoidf_(v00); v01 = sigmoidf_(v01);
            v10 = sigmoidf_(v10); v11 = sigmoidf_(v11);
        } else if (EPI == 1) {
            v00 += bias[(long)rowA * ldc + colA];
            v01 += bias[(long)rowA * ldc + colB];
            v10 += bias[(long)rowB * ldc + colA];
            v11 += bias[(long)rowB * ldc + colB];
        }
        Cb[(long)rowA * ldc + colA] = v00;
        Cb[(long)rowA * ldc + colB] = v01;
        Cb[(long)rowB * ldc + colA] = v10;
        Cb[(long)rowB * ldc + colB] = v11;
    }
}

// row-wise softmax, one block (256 thr) per row
__global__ void softmax_rows(float* __restrict__ data, int ncols) {
    __shared__ float red[256];
    const long row = blockIdx.x;
    float* p = data + row * (long)ncols;
    const int tid = threadIdx.x;
    float m = -3.0e38f;
    for (int c = tid; c < ncols; c += 256) m = fmaxf(m, p[c]);
    red[tid] = m; __syncthreads();
    for (int s = 128; s > 0; s >>= 1) { if (tid < s) red[tid] = fmaxf(red[tid], red[tid + s]); __syncthreads(); }
    m = red[0]; __syncthreads();
    float sum = 0.f;
    for (int c = tid; c < ncols; c += 256) { float e = __expf(p[c] - m); p[c] = e; sum += e; }
    red[tid] = sum; __syncthreads();
    for (int s = 128; s > 0; s >>= 1) { if (tid < s) red[tid] += red[tid + s]; __syncthreads(); }
    const float inv = 1.0f / red[0];
    for (int c = tid; c < ncols; c += 256) p[c] *= inv;
}

// ================= Phase 3: ChebConv =================

__global__ void zero_f32(float* __restrict__ p, long n) {
    const long g = (long)blockIdx.x * 256 + threadIdx.x;
    if (g < n) p[g] = 0.f;
}

__global__ void deg_scatter(const int* __restrict__ row, const int* __restrict__ col,
                            const float* __restrict__ w, float* __restrict__ deg) {
    const int e = blockIdx.x * 256 + threadIdx.x;
    if (e >= EE) return;
    const float wv = (row[e] == col[e]) ? 0.f : w[e];
    atomicAdd(&deg[row[e]], wv);
}

__global__ void compute_dis(const float* __restrict__ deg, float* __restrict__ dis) {
    const int n = blockIdx.x * 256 + threadIdx.x;
    if (n >= NN_) return;
    const float d = deg[n];
    dis[n] = (d > 0.f) ? rsqrtf(d) : 0.f;
}

__global__ void compute_norm(const int* __restrict__ row, const int* __restrict__ col,
                             const float* __restrict__ w, const float* __restrict__ dis,
                             float* __restrict__ norm) {
    const int e = blockIdx.x * 256 + threadIdx.x;
    if (e >= EE) return;
    const float wv = (row[e] == col[e]) ? 0.f : w[e];
    norm[e] = -dis[row[e]] * wv * dis[col[e]];
}

// xg[(b*T+t), n, f] = x_SA[b, n, f, t]
__global__ void transpose_to_xg(const float* __restrict__ xsa, float* __restrict__ xg) {
    const long g = (long)blockIdx.x * 256 + threadIdx.x;   // over BT*N*FIN = 6291456
    const int f = (int)(g & 31);
    const long t1 = g >> 5;
    const int n = (int)(t1 & 1023);
    const long t2 = t1 >> 10;
    const int t = (int)(t2 % TT);
    const int b = (int)(t2 / TT);
    xg[g] = xsa[(long)b * XB + (long)n * D_ + f * TT + t];
}

// out[:,col,:] += scale * norm[e] * z[:,row,:]   (one thread per (e,bt))
__global__ void prop_scatter(const float* __restrict__ z, float* __restrict__ out,
                             const int* __restrict__ row, const int* __restrict__ col,
                             const float* __restrict__ norm, float scale) {
    const long gid = (long)blockIdx.x * 256 + threadIdx.x;   // E*BT
    if (gid >= (long)EE * BT) return;
    const int e  = (int)(gid / BT);
    const int bt = (int)(gid % BT);
    const float nw = norm[e] * scale;
    if (nw == 0.f) return;
    const float* src = z   + ((long)bt * NN_ + row[e]) * FIN;
    float*       dst = out + ((long)bt * NN_ + col[e]) * FIN;
#pragma unroll
    for (int f = 0; f < FIN; ++f) atomicAdd(&dst[f], nw * src[f]);
}

__global__ void axpy_sub(float* __restrict__ y, const float* __restrict__ x, long n) {
    const long g = (long)blockIdx.x * 256 + threadIdx.x;
    if (g < n) y[g] -= x[g];
}

// out[row, fo] (+)= sum_f z[row,f] * W[f,fo]    rows = BT*N, W is 32x64
__global__ void cheb_mm(const float* __restrict__ z, const float* __restrict__ W,
                        float* __restrict__ out, int accumulate) {
    __shared__ float w[FIN * FOUT];
    const int tid = threadIdx.x;
    for (int i = tid; i < FIN * FOUT; i += 256) w[i] = W[i];
    __syncthreads();
    const long gid = (long)blockIdx.x * 256 + tid;   // BT*N*FOUT
    const long rowi = gid >> 6;
    const int fo = (int)(gid & 63);
    const float* zr = z + rowi * FIN;
    float acc = 0.f;
#pragma unroll
    for (int f = 0; f < FIN; ++f) acc += zr[f] * w[f * FOUT + fo];
    if (accumulate) out[gid] += acc; else out[gid] = acc;
}

// ================= Phase 4: residual + relu =================
// out[b,n,fo,t] = relu( gcn[(b*T+t),n,fo] + cheb_b[fo] + res_b[fo] + sum_f resW[fo,f]*x[b,n,f,t] )
__global__ void final_out(const float* __restrict__ gcn, const float* __restrict__ x,
                          const float* __restrict__ resW, const float* __restrict__ resb,
                          const float* __restrict__ chebb, float* __restrict__ out) {
    __shared__ float W[FOUT * FIN];
    const int tid = threadIdx.x;
    for (int i = tid; i < FOUT * FIN; i += 256) W[i] = resW[i];
    __syncthreads();
    const long g = (long)blockIdx.x * 256 + tid;     // B*N*T = 196608
    const int t = (int)(g % TT);
    const long bn = g / TT;
    const int n = (int)(bn % NN_);
    const int b = (int)(bn / NN_);
    float xv[FIN];
    const float* xp = x + (long)b * XB + (long)n * D_;
#pragma unroll
    for (int f = 0; f < FIN; ++f) xv[f] = xp[f * TT + t];
    const float* gp = gcn + ((long)(b * TT + t) * NN_ + n) * FOUT;
    float* op = out + (long)b * (NN_ * FOUT * TT) + (long)n * (FOUT * TT) + t;
    for (int fo = 0; fo < FOUT; ++fo) {
        float s = gp[fo] + chebb[fo] + resb[fo];
#pragma unroll
        for (int f = 0; f < FIN; ++f) s += W[fo * FIN + f] * xv[f];
        op[fo * TT] = fmaxf(s, 0.f);
    }
}

// ================= host =================
extern "C" void kernel_launch(void* const* d_in, const int* in_sizes, int n_in,
                              void* d_out, int out_size, void* d_ws, size_t ws_size,
                              hipStream_t stream) {
    const float* x    = (const float*)d_in[0];
    const int*   erow = (const int*)d_in[1];            // (2,E): row then col
    const int*   ecol = erow + EE;
    const float* ew   = (const float*)d_in[2];
    const float* Ve   = (const float*)d_in[3];
    const float* be   = (const float*)d_in[4];
    const float* Vs   = (const float*)d_in[5];
    const float* bs   = (const float*)d_in[6];
    const float* chW  = (const float*)d_in[7];          // (3,32,64)
    const float* chb  = (const float*)d_in[8];
    const float* rW   = (const float*)d_in[9];
    const float* rb   = (const float*)d_in[10];
    float* out = (float*)d_out;
    float* ws  = (float*)d_ws;

    // workspace layout (floats)
    float* scoreT = ws;                        // 4608
    float* Eatt   = scoreT + 4608;             // 4608
    float* deg    = Eatt + 4608;               // 1024
    float* dis    = deg + 1024;                // 1024
    float* nrm    = dis + 1024;                // 16384
    float* XTA    = ws + 32768;                // B*MFLAT*T = 6291456  (xr; later xg/Tx0)
    float* SIG    = XTA + 6291456;             // B*N*N = 8388608      (sigmoid(score_s); later Tx1)
    float* SAT    = SIG + 8388608;             // B*N*N = 8388608      (S_att; later Tx2)
    float* XSA    = SAT + 8388608;             // 6291456
    float* GCN    = XSA + 6291456;             // BT*N*FOUT = 12582912
    float* Tx0 = XTA;
    float* Tx1 = SIG;
    float* Tx2 = SAT;

    // ---- phase 1: temporal attention ----
    temporal_score<<<BB, 576, 0, stream>>>(x, scoreT);
    temporal_att<<<1, BB * TT, 0, stream>>>(scoreT, Ve, be, Eatt);
    apply_temporal<<<(BB * MFLAT) / 256, 256, 0, stream>>>(x, Eatt, XTA);

    // ---- edge preprocessing (independent) ----
    zero_f32<<<(NN_ + 255) / 256, 256, 0, stream>>>(deg, NN_);
    deg_scatter<<<EE / 256, 256, 0, stream>>>(erow, ecol, ew, deg);
    compute_dis<<<(NN_ + 255) / 256, 256, 0, stream>>>(deg, dis);
    compute_norm<<<EE / 256, 256, 0, stream>>>(erow, ecol, ew, dis, nrm);

    // ---- phase 2: spatial attention (WMMA fp32 GEMMs) ----
    // sig(score_s) = sigmoid(xr @ xr^T)
    wmma_gemm_f32<1, 0><<<dim3(NN_ / 128, NN_ / 32, BB), 128, 0, stream>>>(
        XTA, XTA, SIG, nullptr, D_, D_, NN_, D_, (long)NN_ * D_, (long)NN_ * D_, (long)NN_ * NN_);
    // S_lin = Vs @ sig + bs   (Vs, bs shared across batch)
    wmma_gemm_f32<0, 1><<<dim3(NN_ / 128, NN_ / 32, BB), 128, 0, stream>>>(
        Vs, SIG, SAT, bs, NN_, NN_, NN_, NN_, 0L, (long)NN_ * NN_, (long)NN_ * NN_);
    softmax_rows<<<BB * NN_, 256, 0, stream>>>(SAT, NN_);
    // x_SA = S_att @ xr
    wmma_gemm_f32<0, 2><<<dim3(NN_ / 128, D_ / 32, BB), 128, 0, stream>>>(
        SAT, XTA, XSA, nullptr, NN_, D_, D_, NN_, (long)NN_ * NN_, (long)NN_ * D_, (long)NN_ * D_);

    // ---- phase 3: ChebConv ----
    const long ZN = (long)BT * NN_ * FIN;      // 6291456
    transpose_to_xg<<<(unsigned)(ZN / 256), 256, 0, stream>>>(XSA, Tx0);
    cheb_mm<<<(unsigned)((long)BT * NN_ * FOUT / 256), 256, 0, stream>>>(Tx0, chW + 0 * FIN * FOUT, GCN, 0);
    zero_f32<<<(unsigned)(ZN / 256), 256, 0, stream>>>(Tx1, ZN);
    prop_scatter<<<(unsigned)((long)EE * BT / 256), 256, 0, stream>>>(Tx0, Tx1, erow, ecol, nrm, 1.0f);
    cheb_mm<<<(unsigned)((long)BT * NN_ * FOUT / 256), 256, 0, stream>>>(Tx1, chW + 1 * FIN * FOUT, GCN, 1);
    zero_f32<<<(unsigned)(ZN / 256), 256, 0, stream>>>(Tx2, ZN);
    prop_scatter<<<(unsigned)((long)EE * BT / 256), 256, 0, stream>>>(Tx1, Tx2, erow, ecol, nrm, 2.0f);
    axpy_sub<<<(unsigned)(ZN / 256), 256, 0, stream>>>(Tx2, Tx0, ZN);
    cheb_mm<<<(unsigned)((long)BT * NN_ * FOUT / 256), 256, 0, stream>>>(Tx2, chW + 2 * FIN * FOUT, GCN, 1);

    // ---- phase 4: residual + relu ----
    final_out<<<(BB * NN_ * TT) / 256, 256, 0, stream>>>(GCN, x, rW, rb, chb, out);

    (void)in_sizes; (void)n_in; (void)out_size; (void)ws_size;
}